// LieConvGIGP_44667659878781
// MI455X (gfx1250) — compile-verified
//
#include <hip/hip_runtime.h>
#include <hip/hip_bf16.h>

typedef __attribute__((ext_vector_type(2))) float v2f;
typedef __attribute__((ext_vector_type(8))) float v8f;

#define BS 16
#define NPTS 4096
#define CH 32
#define NORB 64
#define SPLIT 8           // blocks per batch in the streaming phase
#define AGG_ELEMS (NORB * CH)   // 2048 floats per 64x32 tile

// ---------------------------------------------------------------------------
// Phase A: stream vals/coords/mask, aggregate into per-wave LDS tiles,
// reduce waves, emit one 64x32 partial per block into d_ws.
// grid = BS*SPLIT blocks, 256 threads (8 waves); each wave handles 64 points.
// ---------------------------------------------------------------------------
__global__ void __launch_bounds__(256)
gigp_aggregate_kernel(const float* __restrict__ coords,
                      const float* __restrict__ vals,
                      const unsigned char* __restrict__ mask,
                      float* __restrict__ partial) {
    __shared__ float lds[8 * AGG_ELEMS];   // 64 KB: one 64x32 tile per wave

    const int tid  = threadIdx.x;
    const int lane = tid & 31;             // channel
    const int wave = tid >> 5;
    const int b    = blockIdx.x >> 3;      // batch
    const int s    = blockIdx.x & (SPLIT - 1);

    for (int i = tid; i < 8 * AGG_ELEMS; i += 256) lds[i] = 0.0f;
    __syncthreads();

    float* acc = lds + wave * AGG_ELEMS;
    const int pbase = s * (NPTS / SPLIT) + wave * 64;
    #pragma unroll 4
    for (int pp = 0; pp < 64; ++pp) {
        const long idx = (long)b * NPTS + pbase + pp;
        const int  k   = (int)coords[idx * 4 + 3];       // coords[b,p,1,1] = orb id
        const float mv = mask[idx] ? vals[idx * CH + lane] : 0.0f;
        // ds_add_f32, non-returning: pipelined, and deterministic because each
        // LDS address is only ever written by one (wave,lane) in program order.
        atomicAdd(&acc[k * CH + lane], mv);
    }
    __syncthreads();

    float* dst = partial + (long)blockIdx.x * AGG_ELEMS;
    for (int i = tid; i < AGG_ELEMS; i += 256) {
        float t = 0.0f;
        #pragma unroll
        for (int r = 0; r < 8; ++r) t += lds[r * AGG_ELEMS + i];
        dst[i] = t;
    }
}

// ---------------------------------------------------------------------------
// Phase B: per-batch MLP over the 64x32 aggregate using V_WMMA_F32_16X16X4_F32.
// grid = BS blocks, 128 threads (4 waves); wave w owns orbital rows [16w,16w+16).
// ---------------------------------------------------------------------------
__global__ void __launch_bounds__(128)
gigp_mlp_kernel(const float* __restrict__ partial,
                const float* __restrict__ W1, const float* __restrict__ b1,
                const float* __restrict__ W2, const float* __restrict__ b2,
                const float* __restrict__ W3, const float* __restrict__ b3,
                float* __restrict__ out) {
    __shared__ float agg[AGG_ELEMS];       // 64x32 fp32
    __shared__ float hbuf[4 * 512];        // per wave: h1 @ +0 (256), h2 @ +256
    __shared__ float outbuf[NORB];

    const int tid  = threadIdx.x;
    const int lane = tid & 31;
    const int wave = tid >> 5;
    const int b    = blockIdx.x;

    // Deterministic fixed-order reduction of the SPLIT partials.
    for (int i = tid; i < AGG_ELEMS; i += 128) {
        float t = 0.0f;
        #pragma unroll
        for (int s = 0; s < SPLIT; ++s)
            t += partial[((long)b * SPLIT + s) * AGG_ELEMS + i];
        agg[i] = t;
    }
    __syncthreads();

    const int col   = lane & 15;           // N (and M for A-frag rows)
    const int khalf = (lane >> 4) << 1;    // 0 for lanes 0-15, 2 for lanes 16-31
    const int rowhi = (lane >> 4) << 3;    // D-layout: +8 rows for upper half

    // ---- Layer 1: H1 = relu(agg(64x32) @ W1(32x16) + b1), tile M=16 per wave
    v8f c1;
    {
        const float bias = b1[col];
        #pragma unroll
        for (int r = 0; r < 8; ++r) c1[r] = bias;
        #pragma unroll
        for (int kc = 0; kc < 8; ++kc) {
            const int kb = kc * 4 + khalf;
            v2f a, bb;
            a.x  = agg[(wave * 16 + col) * CH + kb];
            a.y  = agg[(wave * 16 + col) * CH + kb + 1];
            bb.x = W1[(kb + 0) * 16 + col];
            bb.y = W1[(kb + 1) * 16 + col];
            c1 = __builtin_amdgcn_wmma_f32_16x16x4_f32(
                    false, a, false, bb, (short)0, c1, false, false);
        }
        float* h1 = hbuf + wave * 512;
        #pragma unroll
        for (int r = 0; r < 8; ++r) {
            const float v = c1[r] > 0.0f ? c1[r] : 0.0f;  // relu
            h1[(r + rowhi) * 16 + col] = v;               // D-layout -> row-major
        }
    }
    __syncthreads();

    // ---- Layer 2: H2 = relu(H1(16x16) @ W2(16x16) + b2)
    v8f c2;
    {
        const float* h1 = hbuf + wave * 512;
        const float bias = b2[col];
        #pragma unroll
        for (int r = 0; r < 8; ++r) c2[r] = bias;
        #pragma unroll
        for (int kc = 0; kc < 4; ++kc) {
            const int kb = kc * 4 + khalf;
            v2f a, bb;
            a.x  = h1[col * 16 + kb];
            a.y  = h1[col * 16 + kb + 1];
            bb.x = W2[(kb + 0) * 16 + col];
            bb.y = W2[(kb + 1) * 16 + col];
            c2 = __builtin_amdgcn_wmma_f32_16x16x4_f32(
                    false, a, false, bb, (short)0, c2, false, false);
        }
        float* h2 = hbuf + wave * 512 + 256;
        #pragma unroll
        for (int r = 0; r < 8; ++r) {
            const float v = c2[r] > 0.0f ? c2[r] : 0.0f;  // relu
            h2[(r + rowhi) * 16 + col] = v;
        }
    }
    __syncthreads();

    // ---- Layer 3: t[m] = H2[m,:] . W3 + b3, zero empty orbitals (agg row-sum == 0)
    if (lane < 16) {
        const float* h2 = hbuf + wave * 512 + 256;
        const int m = lane;
        float rowsum = 0.0f;
        #pragma unroll
        for (int c = 0; c < CH; ++c) rowsum += agg[(wave * 16 + m) * CH + c];
        float t = b3[0];
        #pragma unroll
        for (int n = 0; n < 16; ++n) t += h2[m * 16 + n] * W3[n];
        if (rowsum == 0.0f) t = 0.0f;
        outbuf[wave * 16 + m] = t;
    }
    __syncthreads();

    if (tid == 0) {
        float s = 0.0f;
        #pragma unroll
        for (int i = 0; i < NORB; ++i) s += outbuf[i];
        out[b] = s;
    }
}

extern "C" void kernel_launch(void* const* d_in, const int* in_sizes, int n_in,
                              void* d_out, int out_size, void* d_ws, size_t ws_size,
                              hipStream_t stream) {
    const float*         coords = (const float*)d_in[0];
    const float*         vals   = (const float*)d_in[1];
    const unsigned char* mask   = (const unsigned char*)d_in[2];
    const float*         W1     = (const float*)d_in[3];
    const float*         b1     = (const float*)d_in[4];
    const float*         W2     = (const float*)d_in[5];
    const float*         b2     = (const float*)d_in[6];
    const float*         W3     = (const float*)d_in[7];
    const float*         b3     = (const float*)d_in[8];
    float*               out    = (float*)d_out;

    float* partial = (float*)d_ws;  // BS*SPLIT*64*32 floats = 1 MB

    gigp_aggregate_kernel<<<BS * SPLIT, 256, 0, stream>>>(coords, vals, mask, partial);
    gigp_mlp_kernel<<<BS, 128, 0, stream>>>(partial, W1, b1, W2, b2, W3, b3, out);
}